// CRsAE1d_84284438216823
// MI455X (gfx1250) — compile-verified
//
#include <hip/hip_runtime.h>
#include <math.h>

#define T_ITERS 20
#define KS 31
#define KF 64
#define NCODE 4096
#define NSIG (NCODE - KS + 1)   /* 4066 */
#define BATCH 16

typedef float v2f __attribute__((ext_vector_type(2)));
typedef float v8f __attribute__((ext_vector_type(8)));

/* ---------------- Kernel 1: r = signal - conv_fwd(w),  w = z + mom*(z - zprev)
 * M=1-channel GEMV-like contraction (1984 taps) -> VALU out of LDS.           */
#define FWD_TILE 128
#define FWD_WIN  (FWD_TILE + KS - 1)   /* 158 */
#define FWD_WPAD 160

__global__ __launch_bounds__(FWD_TILE)
void crsae_fwd_kernel(const float* __restrict__ signal,
                      const float* __restrict__ H,
                      const float* __restrict__ zcur,
                      const float* __restrict__ zprev,
                      float* __restrict__ r,
                      float mom)
{
    __shared__ float h_s[KF * KS];         /* ~7.9 KB */
    __shared__ float w_s[KF * FWD_WPAD];   /* 40 KB   */

    const int tid = threadIdx.x;
    const int b   = blockIdx.y;
    const int n0  = blockIdx.x * FWD_TILE;

    for (int i = tid; i < KF * KS; i += FWD_TILE) h_s[i] = H[i];

    const size_t zb = (size_t)b * KF * NCODE;
    for (int i = tid; i < KF * FWD_WIN; i += FWD_TILE) {
        int k   = i / FWD_WIN;
        int col = i - k * FWD_WIN;
        int n   = n0 + col;
        float wv = 0.0f;
        if (n < NCODE) {
            size_t g = zb + (size_t)k * NCODE + n;
            float zc = zcur[g], zp = zprev[g];
            wv = zc + mom * (zc - zp);
        }
        w_s[k * FWD_WPAD + col] = wv;
    }
    __syncthreads();

    const int n = n0 + tid;
    if (n < NSIG) {
        float acc = 0.0f;
        for (int k = 0; k < KF; ++k) {
            const float* hr = &h_s[k * KS];
            const float* wr = &w_s[k * FWD_WPAD + tid];
            #pragma unroll
            for (int j = 0; j < KS; ++j) acc = fmaf(hr[j], wr[j], acc);
        }
        r[(size_t)b * NSIG + n] = signal[(size_t)b * NSIG + n] - acc;
    }
}

/* ---------------- Kernel 2: c = w + (H @ Toeplitz(r))/L ; z_new = shrink(c)
 * True GEMM: C(64 x 4096) = Hpad(64 x 32) @ R(32 x 4096), R[j,m] = r[m-j].
 * fp32 WMMA 16x16x4, 8 back-to-back k-steps per 16x16 tile.
 * r window stored FLIPPED in LDS so each B-operand VGPR pair is an ascending
 * address pair (no swap movs), and all 8 B fragments are preloaded before the
 * WMMA chain (one dscnt wait per tile instead of one per WMMA).               */
#define ADJ_TILE 256
#define KPAD 32
#define RWIN (ADJ_TILE + KS)   /* 287 */

__global__ __launch_bounds__(256)
void crsae_adj_kernel(const float* __restrict__ H,
                      const float* __restrict__ r,
                      const float* __restrict__ zcur,
                      const float* __restrict__ zprev,
                      float* __restrict__ znew,
                      const float* __restrict__ lam_p,
                      const float* __restrict__ L_p,
                      float mom)
{
    __shared__ float hp[KF * KPAD];  /* 8 KB, H zero-padded K=31 -> 32 */
    __shared__ float rsf[RWIN];      /* flipped: rsf[i] = r[m0 + 255 - i] */

    const int tid  = threadIdx.x;
    const int b    = blockIdx.y;
    const int m0   = blockIdx.x * ADJ_TILE;
    const int lane = tid & 31;
    const int wave = tid >> 5;
    const int half = lane >> 4;      /* 0 or 1 */
    const int l15  = lane & 15;

    for (int i = tid; i < KF * KPAD; i += 256) {
        int row = i >> 5, col = i & 31;
        hp[i] = (col < KS) ? H[row * KS + col] : 0.0f;
    }
    for (int i = tid; i < RWIN; i += 256) {
        int g = m0 + (ADJ_TILE - 1) - i;            /* m0+255 down to m0-31 */
        rsf[i] = (g >= 0 && g < NSIG) ? r[(size_t)b * NSIG + g] : 0.0f;
    }
    __syncthreads();

    const float Lv    = L_p[0];
    const float alpha = lam_p[0] / Lv;
    const float invL  = 1.0f / Lv;

    const int kt  = wave & 3;   /* filter-row tile: 4 x 16 = 64 filters */
    const int grp = wave >> 2;  /* column-tile group */

    /* A fragments (H tile) are invariant across column tiles: hoist. */
    v2f afrag[8];
    #pragma unroll
    for (int s = 0; s < 8; ++s) {
        const int ka = s * 4 + 2 * half;            /* K for .x; .y = ka+1 */
        afrag[s].x = hp[(kt * 16 + l15) * KPAD + ka];
        afrag[s].y = hp[(kt * 16 + l15) * KPAD + ka + 1];
    }

    for (int mt8 = 0; mt8 < 8; ++mt8) {
        const int mt   = grp * 8 + mt8;             /* 16 col-tiles of 16 */
        const int colb = mt * 16 + l15;

        /* Preload all B fragments: R[ka, m0+colb] = r[m0+colb-ka]
         * = rsf[255 - colb + ka]; pair (ka, ka+1) -> ascending addresses. */
        v2f bfrag[8];
        #pragma unroll
        for (int s = 0; s < 8; ++s) {
            const int idx = (ADJ_TILE - 1) - colb + s * 4 + 2 * half;
            bfrag[s].x = rsf[idx];
            bfrag[s].y = rsf[idx + 1];
        }

        v8f c = {};
        #pragma unroll
        for (int s = 0; s < 8; ++s)
            c = __builtin_amdgcn_wmma_f32_16x16x4_f32(
                    false, afrag[s], false, bfrag[s], (short)0, c, false, false);

        /* Epilogue: batch loads, then compute+store.
         * D layout: vgpr i, lane -> row kt*16 + i + 8*half, col = colb.   */
        const size_t gbase =
            ((size_t)b * KF + (size_t)(kt * 16 + 8 * half)) * (size_t)NCODE
            + (size_t)(m0 + colb);
        float zc8[8], zp8[8];
        #pragma unroll
        for (int i = 0; i < 8; ++i) zc8[i] = zcur[gbase + (size_t)i * NCODE];
        #pragma unroll
        for (int i = 0; i < 8; ++i) zp8[i] = zprev[gbase + (size_t)i * NCODE];
        #pragma unroll
        for (int i = 0; i < 8; ++i) {
            float wv  = zc8[i] + mom * (zc8[i] - zp8[i]);
            float val = wv + c[i] * invL;
            float av  = fabsf(val) - alpha;
            znew[gbase + (size_t)i * NCODE] = (av > 0.0f) ? copysignf(av, val) : 0.0f;
        }
    }
}

extern "C" void kernel_launch(void* const* d_in, const int* in_sizes, int n_in,
                              void* d_out, int out_size, void* d_ws, size_t ws_size,
                              hipStream_t stream)
{
    (void)in_sizes; (void)n_in; (void)out_size; (void)ws_size;

    const float* signal = (const float*)d_in[0];
    const float* H      = (const float*)d_in[1];
    const float* lam    = (const float*)d_in[2];
    const float* L      = (const float*)d_in[3];

    const size_t Z_BYTES = (size_t)BATCH * KF * NCODE * sizeof(float); /* 16.78 MB */
    float* zA = (float*)d_out;                      /* receives final z (t=19) */
    float* zB = (float*)d_ws;
    float* r  = (float*)((char*)d_ws + Z_BYTES);    /* BATCH*NSIG floats */

    hipMemsetAsync(zA, 0, Z_BYTES, stream);
    hipMemsetAsync(zB, 0, Z_BYTES, stream);

    dim3 g1((NSIG + FWD_TILE - 1) / FWD_TILE, BATCH);  /* 32 x 16 */
    dim3 g2(NCODE / ADJ_TILE, BATCH);                  /* 16 x 16 */

    float* cur   = zA;
    float* other = zB;
    float ps = 0.0f;
    for (int t = 0; t < T_ITERS; ++t) {
        float s   = 0.5f * (1.0f + sqrtf(1.0f + 4.0f * ps * ps));
        float mom = (ps - 1.0f) / s;
        ps = s;
        crsae_fwd_kernel<<<g1, FWD_TILE, 0, stream>>>(signal, H, cur, other, r, mom);
        /* z_new overwrites prev_z in place (each element read-then-written by one thread) */
        crsae_adj_kernel<<<g2, 256, 0, stream>>>(H, r, cur, other, other, lam, L, mom);
        float* tmp = cur; cur = other; other = tmp;
    }
    /* write targets alternate zB,zA,...; t=19 (odd) lands in zA == d_out */
}